// PDESolvedSignature_74088185856481
// MI455X (gfx1250) — compile-verified
//
#include <hip/hip_runtime.h>
#include <hip/hip_bf16.h>
#include <math.h>

// B=16 paths, T=64 timesteps, C=8 channels, f=4 dyadic, N=(T-1)*4=252
#define NB   16
#define NT   64
#define NC   8
#define NN   252          // fine grid size
#define NC1  63           // coarse inc size (T-1)

typedef __attribute__((ext_vector_type(2))) float v2f;
typedef __attribute__((ext_vector_type(8))) float v8f;

// One workgroup per ordered pair (a,b). 256 threads = 8 waves.
__global__ __launch_bounds__(256)
void sigkernel_pair_kernel(const float* __restrict__ paths, float* __restrict__ Kout)
{
    const int pair = blockIdx.x;
    const int a = pair >> 4;
    const int b = pair & 15;
    const int tid = threadIdx.x;

    __shared__ float Xa[NT][NC];        // 2 KB  (X[a][t][c] = paths[a, c, t])
    __shared__ float Xb[NT][NC];        // 2 KB
    __shared__ float sqa[NT];
    __shared__ float sqb[NT];
    __shared__ float G[NT][NT + 1];     // 16.6 KB Gram
    __shared__ float inc[NC1][NT];      // 16 KB coarse increments (pre-scaled 1/16)
    __shared__ float rows[3][256];      // rolling anti-diagonal rows (N+1=253 used)

    // ---- Stage 1: load Xa, Xb (each 512 floats; 2 per thread) ----
    {
        const float* pa = paths + a * (NC * NT);
        const float* pb = paths + b * (NC * NT);
        #pragma unroll
        for (int r = 0; r < 2; ++r) {
            int e = tid + r * 256;        // 0..511
            int c = e >> 6;               // channel
            int t = e & 63;               // time
            Xa[t][c] = pa[c * NT + t];
            Xb[t][c] = pb[c * NT + t];
        }
    }
    __syncthreads();

    // ---- Stage 2: squared norms per timestep ----
    if (tid < NT) {
        float s = 0.f;
        #pragma unroll
        for (int c = 0; c < NC; ++c) s += Xa[tid][c] * Xa[tid][c];
        sqa[tid] = s;
    } else if (tid < 2 * NT) {
        int t = tid - NT;
        float s = 0.f;
        #pragma unroll
        for (int c = 0; c < NC; ++c) s += Xb[t][c] * Xb[t][c];
        sqb[t] = s;
    }
    __syncthreads();

    // ---- Stage 3: ip = Xa @ Xb^T via V_WMMA_F32_16X16X4_F32, then G = exp(-d2) ----
    // 4x4 grid of 16x16 tiles; 8 waves, 2 tiles each; K=8 as two k=4 WMMA steps.
    {
        const int wave = tid >> 5;
        const int lane = tid & 31;
        const int half = lane >> 4;    // 0: lanes 0-15, 1: lanes 16-31
        const int l    = lane & 15;

        #pragma unroll
        for (int tt = 0; tt < 2; ++tt) {
            const int tile = wave * 2 + tt;    // 0..15
            const int m0 = (tile >> 2) * 16;
            const int n0 = (tile & 3) * 16;

            v8f acc = {};
            #pragma unroll
            for (int ks = 0; ks < 2; ++ks) {
                const int k0 = ks * 4;
                // A 16x4 f32: lanes 0-15 hold K=k0+{0,1}; lanes 16-31 hold K=k0+{2,3}
                v2f Af, Bf;
                Af.x = Xa[m0 + l][k0 + 2 * half];
                Af.y = Xa[m0 + l][k0 + 2 * half + 1];
                // B 4x16 f32 (B[k][n] = Xb[n0+n][k]): mirrored striping over lanes
                Bf.x = Xb[n0 + l][k0 + 2 * half];
                Bf.y = Xb[n0 + l][k0 + 2 * half + 1];
                acc = __builtin_amdgcn_wmma_f32_16x16x4_f32(
                        /*neg_a=*/false, Af, /*neg_b=*/false, Bf,
                        /*c_mod=*/(short)0, acc,
                        /*reuse_a=*/false, /*reuse_b=*/false);
            }
            // C/D layout: VGPR r -> M = half*8 + r, N = l
            #pragma unroll
            for (int r = 0; r < 8; ++r) {
                const int m = m0 + half * 8 + r;
                const int n = n0 + l;
                const float d2 = sqa[m] + sqb[n] - 2.0f * acc[r];
                G[m][n] = __expf(-d2);   // SIGMA = 1
            }
        }
    }
    __syncthreads();

    // ---- Stage 4: coarse increments, pre-divided by f*f = 16 ----
    for (int idx = tid; idx < NC1 * NC1; idx += 256) {
        const int p = idx / NC1;
        const int q = idx - p * NC1;
        inc[p][q] = (G[p + 1][q + 1] + G[p][q] - G[p + 1][q] - G[p][q + 1]) * 0.0625f;
    }

    // ---- Stage 5: Goursat PDE anti-diagonal wavefront ----
    rows[0][tid] = 1.0f;
    rows[1][tid] = 1.0f;
    __syncthreads();

    int p2 = 0, p1 = 1, pc = 2;
    const int m = tid;                   // cell index on the diagonal, 0..N
    for (int d = 2; d <= 2 * NN; ++d) {
        if (m <= NN) {
            float val;
            if (m == 0) {
                val = 1.0f;
            } else {
                const int jm1 = d - m - 1;           // fine inc column index
                float Sd = 0.0f;
                if (jm1 >= 0 && jm1 < NN)
                    Sd = inc[(m - 1) >> 2][jm1 >> 2]; // dyadic upsample on the fly
                const float q12 = Sd * Sd * (1.0f / 12.0f);
                const float ab  = rows[p1][m - 1] + rows[p1][m];
                val = ab * (1.0f + 0.5f * Sd + q12) - rows[p2][m - 1] * (1.0f - q12);
            }
            if (m == d) val = 1.0f;                  // boundary U[d,0] = 1
            rows[pc][m] = val;
        }
        __syncthreads();
        const int tmp = p2; p2 = p1; p1 = pc; pc = tmp;
    }

    // final row is rows[p1]; answer K[a,b] = U[N,N]
    if (tid == 0) Kout[a * NB + b] = rows[p1][NN];
}

// Deterministic reduction: out[a] = [K[a,a], mean_b K[a,b]]
__global__ void sigkernel_reduce_kernel(const float* __restrict__ K, float* __restrict__ out)
{
    const int a = threadIdx.x;
    if (a < NB) {
        float s = 0.f;
        #pragma unroll
        for (int bb = 0; bb < NB; ++bb) s += K[a * NB + bb];
        out[2 * a + 0] = K[a * NB + a];
        out[2 * a + 1] = s * (1.0f / NB);
    }
}

extern "C" void kernel_launch(void* const* d_in, const int* in_sizes, int n_in,
                              void* d_out, int out_size, void* d_ws, size_t ws_size,
                              hipStream_t stream) {
    const float* paths = (const float*)d_in[0];   // (16, 8, 64) f32
    float* out = (float*)d_out;                    // 32 f32
    float* Kmat = (float*)d_ws;                    // 256 f32 scratch

    sigkernel_pair_kernel<<<NB * NB, 256, 0, stream>>>(paths, Kmat);
    sigkernel_reduce_kernel<<<1, 32, 0, stream>>>(Kmat, out);
}